// composed_loss_function_35519379538281
// MI455X (gfx1250) — compile-verified
//
#include <hip/hip_runtime.h>

typedef __attribute__((ext_vector_type(2))) float v2f;
typedef __attribute__((ext_vector_type(8))) float v8f;

#define WAVES_PER_BLOCK 8
#define BLOCK_THREADS 256

// sum_n d_n^T C d_n == sum_{m,e} C[m,e] * (D^T D)[m,e]; Gram tiles built with
// V_WMMA_F32_16X16X4_F32 (A and B share one lane layout, so one v2f feeds both).
__global__ void __launch_bounds__(BLOCK_THREADS)
gram_wmma_kernel(const float* __restrict__ pred,   // [N,9]
                 const float* __restrict__ yobs,   // [N,9]
                 const float* __restrict__ rrs,    // [N,5]
                 const float* __restrict__ rrsp,   // [N,5]
                 const float* __restrict__ rcov,   // [5,5]
                 const float* __restrict__ ycov,   // [9,9]
                 float* __restrict__ partials,     // [gridDim.x]
                 int n) {
    const int lane = threadIdx.x & 31;
    const int wave = threadIdx.x >> 5;

    // Scalarize loop control: chunk index/stride are wave-uniform.
    const int gwaveS  = __builtin_amdgcn_readfirstlane(
                            (int)(blockIdx.x * WAVES_PER_BLOCK + wave));
    const int strideS = (int)(gridDim.x * WAVES_PER_BLOCK);
    const int nfull   = n >> 3;   // full 8-row chunks
    const int rem     = n & 7;

    // f32 16x16x4 A-layout: lanes 0-15 carry K=0 (.x), K=1 (.y);
    // lanes 16-31 carry K=2 (.x), K=3 (.y); M (= feature) = lane&15.
    const int   half = lane >> 4;
    const int   col  = lane & 15;
    const int   c9   = col < 9 ? col : 8;
    const float m9   = col < 9 ? 1.0f : 0.0f;
    const int   c5   = col < 5 ? col : 4;
    const float m5   = col < 5 ? 1.0f : 0.0f;

    // Element offsets of this lane's first row within an 8-row chunk.
    const unsigned laneOff9 = (unsigned)(half * 18 + c9);  // (half*2)*9 + c9
    const unsigned laneOff5 = (unsigned)(half * 10 + c5);  // (half*2)*5 + c5

    v8f acc9 = {0.f, 0.f, 0.f, 0.f, 0.f, 0.f, 0.f, 0.f};
    v8f acc5 = {0.f, 0.f, 0.f, 0.f, 0.f, 0.f, 0.f, 0.f};

    for (int ch = gwaveS; ch < nfull; ch += strideS) {
        const unsigned o9 = (unsigned)ch * 72u + laneOff9;
        const unsigned o5 = (unsigned)ch * 40u + laneOff5;

        // rows r0 = ch*8 + half*2 (+1), and the same +4 for the second WMMA pair
        const float p00 = pred[o9],      p01 = pred[o9 + 9];
        const float p10 = pred[o9 + 36], p11 = pred[o9 + 45];
        const float q00 = yobs[o9],      q01 = yobs[o9 + 9];
        const float q10 = yobs[o9 + 36], q11 = yobs[o9 + 45];
        const float r00 = rrs[o5],       r01 = rrs[o5 + 5];
        const float r10 = rrs[o5 + 20],  r11 = rrs[o5 + 25];
        const float s00 = rrsp[o5],      s01 = rrsp[o5 + 5];
        const float s10 = rrsp[o5 + 20], s11 = rrsp[o5 + 25];

        v2f a9a = {(p00 - q00) * m9, (p01 - q01) * m9};
        v2f a9b = {(p10 - q10) * m9, (p11 - q11) * m9};
        v2f a5a = {(r00 - s00) * m5, (r01 - s01) * m5};
        v2f a5b = {(r10 - s10) * m5, (r11 - s11) * m5};

        acc9 = __builtin_amdgcn_wmma_f32_16x16x4_f32(
            false, a9a, false, a9a, (short)0, acc9, false, false);
        acc9 = __builtin_amdgcn_wmma_f32_16x16x4_f32(
            false, a9b, false, a9b, (short)0, acc9, false, false);
        acc5 = __builtin_amdgcn_wmma_f32_16x16x4_f32(
            false, a5a, false, a5a, (short)0, acc5, false, false);
        acc5 = __builtin_amdgcn_wmma_f32_16x16x4_f32(
            false, a5b, false, a5b, (short)0, acc5, false, false);
    }

    // Remainder rows (n % 8): handled once by global wave 0, with row masking.
    if (rem && gwaveS == 0) {
        const int base = nfull * 8;
#pragma unroll
        for (int s = 0; s < 2; ++s) {          // two K=4 row groups
            const int r0 = base + s * 4 + half * 2;
            const int r1 = r0 + 1;
            const float rm0 = r0 < n ? 1.0f : 0.0f;
            const float rm1 = r1 < n ? 1.0f : 0.0f;
            const unsigned rr0 = (unsigned)(r0 < n ? r0 : n - 1);
            const unsigned rr1 = (unsigned)(r1 < n ? r1 : n - 1);
            v2f a9, a5;
            a9.x = (pred[rr0 * 9u + c9] - yobs[rr0 * 9u + c9]) * (m9 * rm0);
            a9.y = (pred[rr1 * 9u + c9] - yobs[rr1 * 9u + c9]) * (m9 * rm1);
            a5.x = (rrs[rr0 * 5u + c5] - rrsp[rr0 * 5u + c5]) * (m5 * rm0);
            a5.y = (rrs[rr1 * 5u + c5] - rrsp[rr1 * 5u + c5]) * (m5 * rm1);
            acc9 = __builtin_amdgcn_wmma_f32_16x16x4_f32(
                false, a9, false, a9, (short)0, acc9, false, false);
            acc5 = __builtin_amdgcn_wmma_f32_16x16x4_f32(
                false, a5, false, a5, (short)0, acc5, false, false);
        }
    }

    // Branchless contraction with cov-inverse matrices.
    // C/D layout: VGPR j holds (M=j, N=col) lanes 0-15, (M=j+8, N=col) lanes 16-31.
    float part = 0.0f;
#pragma unroll
    for (int j = 0; j < 8; ++j) {
        const int   m   = j + half * 8;
        const int   mi9 = m < 9 ? m : 8;
        const float w9  = ycov[mi9 * 9 + c9] * ((m < 9) ? m9 : 0.0f);
        part += acc9[j] * w9;
        const int   mi5 = m < 5 ? m : 4;
        const float w5  = rcov[mi5 * 5 + c5] * ((m < 5) ? m5 : 0.0f);
        part += acc5[j] * w5;
    }

    // wave32 butterfly reduction
#pragma unroll
    for (int off = 16; off > 0; off >>= 1)
        part += __shfl_xor(part, off, 32);

    __shared__ float wsum[WAVES_PER_BLOCK];
    if (lane == 0) wsum[wave] = part;
    __syncthreads();
    if (threadIdx.x == 0) {
        float s = 0.0f;
#pragma unroll
        for (int i = 0; i < WAVES_PER_BLOCK; ++i) s += wsum[i];
        partials[blockIdx.x] = s;
    }
}

// Deterministic single-block finalize: fixed-order double accumulation,
// plus 0.001 * mean(param[0:3]).
__global__ void __launch_bounds__(BLOCK_THREADS)
finalize_kernel(const float* __restrict__ partials, int nblocks,
                const float* __restrict__ param, float* __restrict__ out) {
    __shared__ double sh[BLOCK_THREADS];
    double s = 0.0;
    for (int i = threadIdx.x; i < nblocks; i += BLOCK_THREADS)
        s += (double)partials[i];
    sh[threadIdx.x] = s;
    __syncthreads();
#pragma unroll
    for (int w = BLOCK_THREADS / 2; w > 0; w >>= 1) {
        if ((int)threadIdx.x < w) sh[threadIdx.x] += sh[threadIdx.x + w];
        __syncthreads();
    }
    if (threadIdx.x == 0) {
        const double pm =
            ((double)param[0] + (double)param[1] + (double)param[2]) / 3.0;
        out[0] = (float)(sh[0] + 0.001 * pm);
    }
}

extern "C" void kernel_launch(void* const* d_in, const int* in_sizes, int n_in,
                              void* d_out, int out_size, void* d_ws, size_t ws_size,
                              hipStream_t stream) {
    const float* pred  = (const float*)d_in[0];  // [N,9]
    const float* yobs  = (const float*)d_in[1];  // [N,9]
    const float* rrs   = (const float*)d_in[2];  // [N,5]
    const float* rrsp  = (const float*)d_in[3];  // [N,5]
    const float* param = (const float*)d_in[5];  // [3]
    const float* rcov  = (const float*)d_in[6];  // [5,5]
    const float* ycov  = (const float*)d_in[7];  // [9,9]

    const int n = in_sizes[0] / 9;

    int nblocks = 768;                           // 6144 waves to saturate HBM
    const int maxb = (int)(ws_size / sizeof(float));
    if (nblocks > maxb) nblocks = maxb;
    const long nfull = (long)(n >> 3);
    const long maxNeeded = (nfull + WAVES_PER_BLOCK - 1) / WAVES_PER_BLOCK;
    if (maxNeeded >= 1 && (long)nblocks > maxNeeded) nblocks = (int)maxNeeded;
    if (nblocks < 1) nblocks = 1;

    gram_wmma_kernel<<<nblocks, BLOCK_THREADS, 0, stream>>>(
        pred, yobs, rrs, rrsp, rcov, ycov, (float*)d_ws, n);
    finalize_kernel<<<1, BLOCK_THREADS, 0, stream>>>(
        (const float*)d_ws, nblocks, param, (float*)d_out);
}